// SumLayerExplicit_81827716923473
// MI455X (gfx1250) — compile-verified
//
#include <hip/hip_runtime.h>
#include <hip/hip_bf16.h>
#include <math.h>

// out[s] = sum_{i+j=s} softmax(d0)[i] * softmax(d1)[j],  s = 0..18
//
// MI455X (gfx1250, wave32) strategy:
//  * single wave, single WGP — ~400 flops / 160 bytes, pure latency problem
//  * outer product p0 (x) p1 in ONE v_wmma_f32_16x16x4_f32 (exact fp32,
//    matches reference numerics; 10x10 live region zero-padded to 16x16)
//  * anti-diagonal reduction via a zero-padded LDS tile: fully unrolled,
//    branch-free, 16 independent ds_loads -> single s_wait_dscnt -> add tree
//    (previous version compiled to a divergent loop with a ds_load +
//    s_wait_dscnt round-trip per iteration — the dominant latency chain)

typedef __attribute__((ext_vector_type(2))) float v2f;
typedef __attribute__((ext_vector_type(8))) float v8f;

#define N_CLASSES 10
#define OUT_N (2 * N_CLASSES - 1)  // 19
#define PADW 48                    // C tile lives in columns [16, 32)

__global__ __launch_bounds__(32)
void sumlayer_wmma_kernel(const float* __restrict__ d0,
                          const float* __restrict__ d1,
                          float* __restrict__ out) {
    __shared__ float s_x0[N_CLASSES];
    __shared__ float s_x1[N_CLASSES];
    __shared__ float s_p0[16];
    __shared__ float s_p1[16];
    __shared__ float s_Cp[16][PADW];  // zero-padded: cols 0..15 and 32..47 = 0

    const int lane = threadIdx.x;  // 0..31, wave32

    // Stage raw logits into LDS so every lane can do the softmax scan.
    if (lane < N_CLASSES) {
        s_x0[lane] = d0[lane];
        s_x1[lane] = d1[lane];
    }

    // Zero the pad columns: lanes 0-15 clear cols 0..15, lanes 16-31 clear
    // cols 32..47, for all 16 rows. Independent of the logit staging above.
    {
        const int pc = (lane < 16) ? lane : (lane + 16);
        #pragma unroll
        for (int m = 0; m < 16; ++m) {
            s_Cp[m][pc] = 0.f;
        }
    }
    __syncthreads();

    // Lanes 0..15 compute their class probability (0 for padded classes
    // 10..15). 10 exps per lane — trivial at this size.
    if (lane < 16) {
        float m0 = -INFINITY, m1 = -INFINITY;
        #pragma unroll
        for (int j = 0; j < N_CLASSES; ++j) {
            m0 = fmaxf(m0, s_x0[j]);
            m1 = fmaxf(m1, s_x1[j]);
        }
        float z0 = 0.f, z1 = 0.f;
        #pragma unroll
        for (int j = 0; j < N_CLASSES; ++j) {
            z0 += __expf(s_x0[j] - m0);
            z1 += __expf(s_x1[j] - m1);
        }
        s_p0[lane] = (lane < N_CLASSES) ? __expf(s_x0[lane] - m0) / z0 : 0.f;
        s_p1[lane] = (lane < N_CLASSES) ? __expf(s_x1[lane] - m1) / z1 : 0.f;
    }
    __syncthreads();

    // Build WMMA fragments for D = A(16x4) * B(4x16) + 0 in fp32, K=0 only.
    // A layout (ISA 7.12.2, 32-bit A 16x4): lanes 0-15 VGPR0 = K=0 (M=lane);
    // lanes 16-31 VGPR0 = K=2. B (4x16): lanes 0-15 VGPR0 = K=0 row, N=lane.
    // Branch-free build: clamped LDS read + select (no exec divergence).
    const float t0 = s_p0[lane & 15];
    const float t1 = s_p1[lane & 15];
    const bool hi = (lane & 16) != 0;
    v2f a = {hi ? 0.f : t0, 0.f};   // A[M=lane][K=0] = p0[lane], K=1..3 = 0
    v2f b = {hi ? 0.f : t1, 0.f};   // B[K=0][N=lane] = p1[lane]

    // WMMA requires EXEC all-1s: we are in uniform control flow here.
    v8f c = {};
    c = __builtin_amdgcn_wmma_f32_16x16x4_f32(
        /*neg_a=*/false, a, /*neg_b=*/false, b,
        /*c_mod=*/(short)0, c, /*reuse_a=*/false, /*reuse_b=*/false);

    // Spill C into the padded tile per the 16x16 f32 C/D layout:
    // lane<16: M = v,   N = lane;   lane>=16: M = v+8, N = lane-16.
    const int n = lane & 15;
    const int mbase = (lane >> 4) * 8;
    #pragma unroll
    for (int v = 0; v < 8; ++v) {
        s_Cp[mbase + v][16 + n] = c[v];
    }
    __syncthreads();

    // Anti-diagonal reduction, fully unrolled and branch-free:
    //   out[s] = sum_{i=0}^{15} Cpad[i][16 + s - i]
    // Column index range for s in [0,18]: [16-15, 16+18] = [1, 34] in [0,48).
    // Pad columns are zero, so no bounds logic needed. 16 independent
    // ds_loads -> one wait -> add tree.
    if (lane < OUT_N) {
        float acc = 0.f;
        #pragma unroll
        for (int i = 0; i < 16; ++i) {
            acc += s_Cp[i][16 + lane - i];
        }
        out[lane] = acc;
    }
}

extern "C" void kernel_launch(void* const* d_in, const int* in_sizes, int n_in,
                              void* d_out, int out_size, void* d_ws, size_t ws_size,
                              hipStream_t stream) {
    (void)in_sizes; (void)n_in; (void)d_ws; (void)ws_size; (void)out_size;
    const float* d0 = (const float*)d_in[0];
    const float* d1 = (const float*)d_in[1];
    float* out = (float*)d_out;
    // Single wave32 — the whole problem fits in one wave on one WGP.
    sumlayer_wmma_kernel<<<1, 32, 0, stream>>>(d0, d1, out);
}